// EngramMemory_8074538517157
// MI455X (gfx1250) — compile-verified
//
#include <hip/hip_runtime.h>
#include <cstdint>

// EngramMemory: hashed n-gram embedding gather.
//   tokens: (4,4096) int      -> d_in[0]
//   tables: (8,2048,256) f32  -> d_in[1]
//   out:    (4,4096,2048) f32 -> d_out
// Pure gather: HBM-write-bound (134 MB out ≈ 5.8 us floor at 23.3 TB/s;
// 16 MB of tables stay resident in the 192 MB L2, so reads are L2 hits).

typedef __attribute__((ext_vector_type(4))) float v4f;
// Match the builtin's parameter type exactly (from hipcc diagnostic):
// 'int __attribute__((vector_size(16))) __device__ *'  == AS(1) v4i*
typedef int v4i __attribute__((__vector_size__(16)));
typedef __attribute__((address_space(1))) v4i gv4i;   // global AS
typedef __attribute__((address_space(3))) v4i lv4i;   // LDS AS

#define ENG_B      4
#define ENG_S      4096
#define ENG_NB     2048u
#define ENG_PRIME  1000003u

// ---- CDNA5 async global<->LDS path selection -------------------------------
#if defined(__HIP_DEVICE_COMPILE__) && defined(__gfx1250__)
  #if __has_builtin(__builtin_amdgcn_global_load_async_to_lds_b128) && \
      __has_builtin(__builtin_amdgcn_global_store_async_from_lds_b128)
    #define ENG_ASYNC_BUILTIN 1
  #else
    #define ENG_ASYNC_ASM 1
  #endif
#endif

__device__ __forceinline__ void eng_wait_async0() {
#if defined(__HIP_DEVICE_COMPILE__) && defined(__gfx1250__)
  #if __has_builtin(__builtin_amdgcn_s_wait_asynccnt)
    __builtin_amdgcn_s_wait_asynccnt(0);
  #else
    asm volatile("s_wait_asynccnt 0" ::: "memory");
  #endif
#endif
}

__global__ __launch_bounds__(256) void engram_gather_kernel(
    const int* __restrict__ tokens,
    const float* __restrict__ tables,
    float* __restrict__ out)
{
  const uint32_t token_id = blockIdx.x;          // b*4096 + s
  const uint32_t s   = token_id & (ENG_S - 1u);
  const uint32_t tid = threadIdx.x;              // 0..255

  // Uniform (per-block) token fetches -> scalar loads.
  const uint32_t t0 = (uint32_t)tokens[token_id];
  const uint32_t t1 = (s >= 1u) ? (uint32_t)tokens[token_id - 1u] : 0u;
  const uint32_t t2 = (s >= 2u) ? (uint32_t)tokens[token_id - 2u] : 0u;

  // Chunk tid     -> order-2 table g = head       (floats [g*256 + d4*4 ..))
  // Chunk tid+256 -> order-3 table g = head + 4
  const uint32_t head = (tid >> 6) & 3u;
  const uint32_t d4   = tid & 63u;

  // h = (h*PRIME + tok + seed) % 2048 ; 2048 is 2^11 so '%'=='&2047',
  // and 2047*1000003 + 50257 + seed < 2^31 -> u32 math is exact.
  const uint32_t seed2 = 1337u + 97u * 2u + 17u * head;
  uint32_t b2 = (t1 + seed2) & (ENG_NB - 1u);
  b2 = (b2 * ENG_PRIME + t0 + seed2) & (ENG_NB - 1u);

  const uint32_t seed3 = 1337u + 97u * 3u + 17u * head;
  uint32_t b3 = (t2 + seed3) & (ENG_NB - 1u);
  b3 = (b3 * ENG_PRIME + t1 + seed3) & (ENG_NB - 1u);
  b3 = (b3 * ENG_PRIME + t0 + seed3) & (ENG_NB - 1u);

  const v4f* tab  = (const v4f*)tables;
  const v4f* src2 = tab + ((size_t)head         * ENG_NB + b2) * 64u + d4;
  const v4f* src3 = tab + ((size_t)(head + 4u)  * ENG_NB + b3) * 64u + d4;
  v4f* dst = (v4f*)out + (size_t)token_id * 512u;

#if defined(ENG_ASYNC_BUILTIN)
  // Async gather into LDS staging, then async stream LDS -> HBM.
  // Data never passes through VGPRs; tracked by ASYNCcnt.
  __shared__ v4f stage[512];
  lv4i* la = (lv4i*)(uint32_t)(uintptr_t)&stage[tid];
  lv4i* lb = (lv4i*)(uint32_t)(uintptr_t)&stage[tid + 256u];
  __builtin_amdgcn_global_load_async_to_lds_b128((gv4i*)(uintptr_t)src2, la, 0, 0);
  __builtin_amdgcn_global_load_async_to_lds_b128((gv4i*)(uintptr_t)src3, lb, 0, 0);
  eng_wait_async0();  // LDS writes complete
  __builtin_amdgcn_global_store_async_from_lds_b128((gv4i*)(uintptr_t)(dst + tid),        la, 0, 0);
  __builtin_amdgcn_global_store_async_from_lds_b128((gv4i*)(uintptr_t)(dst + tid + 256u), lb, 0, 0);
  eng_wait_async0();  // drain before endpgm
#elif defined(ENG_ASYNC_ASM)
  __shared__ v4f stage[512];
  const uint32_t la = (uint32_t)(uintptr_t)&stage[tid];
  const uint32_t lb = (uint32_t)(uintptr_t)&stage[tid + 256u];
  asm volatile("global_load_async_to_lds_b128 %0, %1, off"
               :: "v"(la), "v"(src2) : "memory");
  asm volatile("global_load_async_to_lds_b128 %0, %1, off"
               :: "v"(lb), "v"(src3) : "memory");
  eng_wait_async0();
  asm volatile("global_store_async_from_lds_b128 %0, %1, off"
               :: "v"(dst + tid), "v"(la) : "memory");
  asm volatile("global_store_async_from_lds_b128 %0, %1, off"
               :: "v"(dst + tid + 256u), "v"(lb) : "memory");
  eng_wait_async0();
#else
  // Portable fallback: B128 load (L2 hit) -> nontemporal B128 store.
  v4f a = src2[0];
  v4f b = src3[0];
  __builtin_nontemporal_store(a, dst + tid);
  __builtin_nontemporal_store(b, dst + tid + 256u);
#endif
}

extern "C" void kernel_launch(void* const* d_in, const int* in_sizes, int n_in,
                              void* d_out, int out_size, void* d_ws, size_t ws_size,
                              hipStream_t stream) {
  (void)in_sizes; (void)n_in; (void)out_size; (void)d_ws; (void)ws_size;
  const int*   tokens = (const int*)d_in[0];
  const float* tables = (const float*)d_in[1];
  float*       out    = (float*)d_out;

  dim3 grid(ENG_B * ENG_S);   // one block per token: 16384 blocks
  dim3 block(256);            // 8 wave32s; each wave moves 512B-coalesced tiles
  engram_gather_kernel<<<grid, block, 0, stream>>>(tokens, tables, out);
}